// GAT_60851096649750
// MI455X (gfx1250) — compile-verified
//
#include <hip/hip_runtime.h>
#include <math.h>

// ---------------------------------------------------------------------------
// Types for CDNA5 WMMA (fp32 path: V_WMMA_F32_16X16X4_F32)
// ---------------------------------------------------------------------------
typedef float v2f __attribute__((ext_vector_type(2)));
typedef float v8f __attribute__((ext_vector_type(8)));

#define NEG_SLOPE 0.2f

// Monotonic uint encoding of float for atomicMax-based segment max.
__device__ __forceinline__ unsigned f2ord(float f) {
    unsigned u = __float_as_uint(f);
    return (u & 0x80000000u) ? ~u : (u | 0x80000000u);
}
__device__ __forceinline__ float ord2f(unsigned u) {
    return (u & 0x80000000u) ? __uint_as_float(u & 0x7FFFFFFFu)
                             : __uint_as_float(~u);
}

// ---------------------------------------------------------------------------
// WMMA fp32 GEMM: Cout[M x Nout] = A[M x K] @ W[Nout x K]^T (+ optional bias)
// One wave: 16 rows x 64 cols (4 col tiles), A fragment reused 4x.
// Branchless OOB handling: load indices are CLAMPED (loop-invariant cndmask),
// correctness preserved because the corresponding D stores are guarded.
// K unrolled by 8 (2 WMMA k-steps): 10 loads issued, then 8 WMMAs.
// Requires K % 8 == 0 (K = 128 or 64 here).
// ---------------------------------------------------------------------------
__global__ void __launch_bounds__(128)
gemm_wmma_f32(const float* __restrict__ A, const float* __restrict__ W,
              const float* __restrict__ bias, float* __restrict__ Cout,
              int M, int K, int Nout) {
    const int wave = threadIdx.x >> 5;
    const int lane = threadIdx.x & 31;
    const int half = lane >> 4;        // 0: K pair {k0,k0+1}, 1: {k0+2,k0+3}
    const int l    = lane & 15;
    const int m0   = (blockIdx.x * 4 + wave) * 16;
    const int n0   = blockIdx.y * 64;

    if (m0 >= M) return;               // wave-uniform: EXEC stays all-1s

    const int arow  = m0 + l;
    const int arowC = arow < M ? arow : (M - 1);          // clamped (loads only)
    const float* Ap = A + (size_t)arowC * K + half * 2;

    const float* Wp[4];
#pragma unroll
    for (int j = 0; j < 4; ++j) {
        const int ncol  = n0 + j * 16 + l;
        const int ncolC = ncol < Nout ? ncol : (Nout - 1); // clamped (loads only)
        Wp[j] = W + (size_t)ncolC * K + half * 2;
    }

    v8f acc[4] = {};

    for (int k0 = 0; k0 < K; k0 += 8) {
        const v2f a0 = *reinterpret_cast<const v2f*>(Ap + k0);
        const v2f a1 = *reinterpret_cast<const v2f*>(Ap + k0 + 4);
        v2f b0[4], b1[4];
#pragma unroll
        for (int j = 0; j < 4; ++j) {
            b0[j] = *reinterpret_cast<const v2f*>(Wp[j] + k0);
            b1[j] = *reinterpret_cast<const v2f*>(Wp[j] + k0 + 4);
        }
#pragma unroll
        for (int j = 0; j < 4; ++j)
            acc[j] = __builtin_amdgcn_wmma_f32_16x16x4_f32(
                false, a0, false, b0[j], (short)0, acc[j], false, false);
#pragma unroll
        for (int j = 0; j < 4; ++j)
            acc[j] = __builtin_amdgcn_wmma_f32_16x16x4_f32(
                false, a1, false, b1[j], (short)0, acc[j], false, false);
    }

#pragma unroll
    for (int j = 0; j < 4; ++j) {
        const int col = n0 + j * 16 + l;
        if (col >= Nout) continue;
        const float bv = bias ? bias[col] : 0.f;
#pragma unroll
        for (int i = 0; i < 8; ++i) {
            const int row = m0 + half * 8 + i;
            if (row < M)
                Cout[(size_t)row * Nout + col] = acc[j][i] + bv;
        }
    }
}

// ---------------------------------------------------------------------------
// Attention coefficients: out[n*8+h] = dot(h[n, head, :], avec[head, :])
// ---------------------------------------------------------------------------
__global__ void attn_k(const float* __restrict__ h, const float* __restrict__ avec,
                       float* __restrict__ out, int Nn, int C) {
    const int t = blockIdx.x * blockDim.x + threadIdx.x;
    if (t >= Nn * 8) return;
    const int n = t >> 3, hh = t & 7;
    const float* hp = h + (size_t)n * 8 * C + (size_t)hh * C;
    const float* ap = avec + (size_t)hh * C;
    float acc = 0.f;
    for (int c = 0; c < C; ++c) acc += hp[c] * ap[c];
    out[t] = acc;
}

// ---------------------------------------------------------------------------
// Per-layer init: acc = 0, m = ord(-inf), s = 0
// ---------------------------------------------------------------------------
__global__ void init_k(float* __restrict__ acc, unsigned* __restrict__ mb,
                       float* __restrict__ sb, int Nn, int HC) {
    const int i = blockIdx.x * blockDim.x + threadIdx.x;
    if (i < Nn * HC) acc[i] = 0.f;
    if (i < Nn * 8) {
        mb[i] = f2ord(-3.0e38f);
        sb[i] = 0.f;
    }
}

// ---------------------------------------------------------------------------
// Edge pass 1: segment max of leaky_relu(als[src] + ald[dst]) per head.
// Edges are [E real edges] ++ [N self loops].
// ---------------------------------------------------------------------------
__global__ void edge_max_k(const int* __restrict__ eidx, int E, int Nn,
                           const float* __restrict__ als,
                           const float* __restrict__ ald,
                           unsigned* __restrict__ mb) {
    const int i = blockIdx.x * blockDim.x + threadIdx.x;
    if (i >= E + Nn) return;
    const int s = (i < E) ? eidx[i] : (i - E);
    const int d = (i < E) ? eidx[E + i] : (i - E);
    const float4* ps = reinterpret_cast<const float4*>(als + (size_t)s * 8);
    const float4* pd = reinterpret_cast<const float4*>(ald + (size_t)d * 8);
    const float4 s0 = ps[0], s1 = ps[1], d0 = pd[0], d1 = pd[1];
    float ev[8] = {s0.x + d0.x, s0.y + d0.y, s0.z + d0.z, s0.w + d0.w,
                   s1.x + d1.x, s1.y + d1.y, s1.z + d1.z, s1.w + d1.w};
#pragma unroll
    for (int hh = 0; hh < 8; ++hh) {
        float e = ev[hh];
        e = e > 0.f ? e : NEG_SLOPE * e;
        atomicMax(mb + (size_t)d * 8 + hh, f2ord(e));
    }
}

// ---------------------------------------------------------------------------
// Edge pass 2: a = exp(e - m[dst]); s[dst] += a; acc[dst] += a * h[src]
// (normalization by s deferred to the node-level finalize)
// One thread per (edge, head); C contiguous atomic f32 adds per thread.
// Templated so gather + atomic loops fully unroll (C = 8 or 40).
// ---------------------------------------------------------------------------
template <int C>
__global__ void edge_acc_k(const int* __restrict__ eidx, int E, int Nn,
                           const float* __restrict__ als,
                           const float* __restrict__ ald,
                           const unsigned* __restrict__ mb,
                           float* __restrict__ sb,
                           const float* __restrict__ h,
                           float* __restrict__ acc) {
    const long t = (long)blockIdx.x * blockDim.x + threadIdx.x;
    if (t >= (long)(E + Nn) * 8) return;
    const int i  = (int)(t >> 3);
    const int hh = (int)(t & 7);
    const int s = (i < E) ? eidx[i] : (i - E);
    const int d = (i < E) ? eidx[E + i] : (i - E);
    float e = als[(size_t)s * 8 + hh] + ald[(size_t)d * 8 + hh];
    e = e > 0.f ? e : NEG_SLOPE * e;
    const float m = ord2f(mb[(size_t)d * 8 + hh]);
    const float a = __expf(e - m);
    atomicAdd(sb + (size_t)d * 8 + hh, a);
    const float* hs = h + (size_t)s * (8 * C) + (size_t)hh * C;
    float* ad = acc + (size_t)d * (8 * C) + (size_t)hh * C;
    float hv[C];
#pragma unroll
    for (int c = 0; c < C; ++c) hv[c] = hs[c];
#pragma unroll
    for (int c = 0; c < C; ++c) atomicAdd(ad + c, a * hv[c]);
}

// ---------------------------------------------------------------------------
// Finalize layers 0/1: feat = elu(acc/(s+eps) + b + skip)   [N x 64]
// ---------------------------------------------------------------------------
__global__ void finalize01_k(const float* __restrict__ acc,
                             const float* __restrict__ sb,
                             const float* __restrict__ b,
                             const float* __restrict__ skip,
                             float* __restrict__ feat, int Nn) {
    const int i = blockIdx.x * blockDim.x + threadIdx.x;
    if (i >= Nn * 64) return;
    const int n = i >> 6, j = i & 63, hh = j >> 3;
    float v = acc[i] / (sb[(size_t)n * 8 + hh] + 1e-16f) + b[j] + skip[i];
    feat[i] = v > 0.f ? v : (__expf(v) - 1.f);  // ELU(alpha=1)
}

// ---------------------------------------------------------------------------
// Finalize layer 2: mean over heads + b + skip, then log_softmax over 40.
// ---------------------------------------------------------------------------
__global__ void finalize2_k(const float* __restrict__ acc,
                            const float* __restrict__ sb,
                            const float* __restrict__ b2,
                            const float* __restrict__ skip,
                            float* __restrict__ out, int Nn) {
    const int n = blockIdx.x * blockDim.x + threadIdx.x;
    if (n >= Nn) return;
    float v[40];
    float inv_s[8];
#pragma unroll
    for (int hh = 0; hh < 8; ++hh)
        inv_s[hh] = 1.f / (sb[(size_t)n * 8 + hh] + 1e-16f);
    for (int c = 0; c < 40; ++c) {
        float sum = 0.f;
#pragma unroll
        for (int hh = 0; hh < 8; ++hh)
            sum += acc[(size_t)n * 320 + hh * 40 + c] * inv_s[hh];
        v[c] = sum * 0.125f + b2[c] + skip[(size_t)n * 40 + c];
    }
    float mx = v[0];
    for (int c = 1; c < 40; ++c) mx = fmaxf(mx, v[c]);
    float se = 0.f;
    for (int c = 0; c < 40; ++c) se += expf(v[c] - mx);
    const float lse = mx + logf(se);
    for (int c = 0; c < 40; ++c) out[(size_t)n * 40 + c] = v[c] - lse;
}

// ---------------------------------------------------------------------------
// Host driver
// ---------------------------------------------------------------------------
extern "C" void kernel_launch(void* const* d_in, const int* in_sizes, int n_in,
                              void* d_out, int out_size, void* d_ws, size_t ws_size,
                              hipStream_t stream) {
    const float* x    = (const float*)d_in[0];
    const int*   eidx = (const int*)d_in[1];
    const float* W0  = (const float*)d_in[2];
    const float* as0 = (const float*)d_in[3];
    const float* ad0 = (const float*)d_in[4];
    const float* b0  = (const float*)d_in[5];
    const float* sW0 = (const float*)d_in[6];
    const float* sb0 = (const float*)d_in[7];
    const float* W1  = (const float*)d_in[8];
    const float* as1 = (const float*)d_in[9];
    const float* ad1 = (const float*)d_in[10];
    const float* b1  = (const float*)d_in[11];
    const float* sW1 = (const float*)d_in[12];
    const float* sb1 = (const float*)d_in[13];
    const float* W2  = (const float*)d_in[14];
    const float* as2 = (const float*)d_in[15];
    const float* ad2 = (const float*)d_in[16];
    const float* b2  = (const float*)d_in[17];
    const float* sW2 = (const float*)d_in[18];
    const float* sb2 = (const float*)d_in[19];

    const int Nn = in_sizes[0] / 128;
    const int E  = in_sizes[1] / 2;

    // Workspace carve-out (~320 MB total)
    float* ws    = (float*)d_ws;
    float* feat  = ws;                            // N*64
    float* hbuf  = feat  + (size_t)Nn * 64;       // N*320
    float* accb  = hbuf  + (size_t)Nn * 320;      // N*320
    float* skipb = accb  + (size_t)Nn * 320;      // N*64
    float* als   = skipb + (size_t)Nn * 64;       // N*8
    float* ald   = als   + (size_t)Nn * 8;        // N*8
    unsigned* mb = (unsigned*)(ald + (size_t)Nn * 8); // N*8
    float* sbf   = (float*)(mb + (size_t)Nn * 8); // N*8

    auto layer = [&](const float* inPtr, int inC, int C,
                     const float* W, const float* aS, const float* aD,
                     const float* b, const float* sW, const float* sbias,
                     int skipC, float* outFeat, bool last) {
        const int HC = 8 * C;
        // Dense WMMA GEMMs: main transform + skip connection
        dim3 gg((Nn + 63) / 64, (HC + 63) / 64);
        gemm_wmma_f32<<<gg, 128, 0, stream>>>(inPtr, W, nullptr, hbuf, Nn, inC, HC);
        dim3 gs((Nn + 63) / 64, (skipC + 63) / 64);
        gemm_wmma_f32<<<gs, 128, 0, stream>>>(inPtr, sW, sbias, skipb, Nn, inC, skipC);
        // Attention coefficients
        const int nh = Nn * 8;
        attn_k<<<(nh + 255) / 256, 256, 0, stream>>>(hbuf, aS, als, Nn, C);
        attn_k<<<(nh + 255) / 256, 256, 0, stream>>>(hbuf, aD, ald, Nn, C);
        // Init per-layer stats + accumulator
        const int na = Nn * HC;
        init_k<<<(na + 255) / 256, 256, 0, stream>>>(accb, mb, sbf, Nn, HC);
        // Edge phase
        const int tot = E + Nn;
        edge_max_k<<<(tot + 255) / 256, 256, 0, stream>>>(eidx, E, Nn, als, ald, mb);
        const long toth = (long)tot * 8;
        const unsigned gb = (unsigned)((toth + 255) / 256);
        if (C == 8)
            edge_acc_k<8><<<gb, 256, 0, stream>>>(eidx, E, Nn, als, ald, mb, sbf,
                                                  hbuf, accb);
        else
            edge_acc_k<40><<<gb, 256, 0, stream>>>(eidx, E, Nn, als, ald, mb, sbf,
                                                   hbuf, accb);
        // Finalize
        if (!last)
            finalize01_k<<<(Nn * 64 + 255) / 256, 256, 0, stream>>>(
                accb, sbf, b, skipb, outFeat, Nn);
        else
            finalize2_k<<<(Nn + 255) / 256, 256, 0, stream>>>(
                accb, sbf, b, skipb, (float*)d_out, Nn);
    };

    layer(x,    128, 8,  W0, as0, ad0, b0, sW0, sb0, 64, feat, false);
    layer(feat, 64,  8,  W1, as1, ad1, b1, sW1, sb1, 64, feat, false);
    layer(feat, 64,  40, W2, as2, ad2, b2, sW2, sb2, 40, nullptr, true);
}